// JSD_6786048328246
// MI455X (gfx1250) — compile-verified
//
#include <hip/hip_runtime.h>
#include <stdint.h>

#define VDIM   32000
#define V4     (VDIM / 4)       // 8000 float4 per row
#define NROWS  4096             // B*S = 2*2048
#define NTHR   1024             // 32 wave32 waves per workgroup
#define NBLK   1024             // grid-stride: each block handles 4 rows

typedef int v4i_t __attribute__((vector_size(16)));

// ---------------- CDNA5 async global->LDS staging -------------------------
__device__ __forceinline__ void async_copy_b128(const void* g, void* l) {
#if __has_builtin(__builtin_amdgcn_global_load_async_to_lds_b128)
  __builtin_amdgcn_global_load_async_to_lds_b128(
      (v4i_t*)g, (v4i_t*)l, 0, 0);
#else
  const uint32_t lds = (uint32_t)(uintptr_t)l;
  const uint64_t ga  = (uint64_t)(uintptr_t)g;
  asm volatile("global_load_async_to_lds_b128 %0, %1, off"
               :: "v"(lds), "v"(ga) : "memory");
#endif
}

__device__ __forceinline__ void wait_async_zero() {
#if __has_builtin(__builtin_amdgcn_s_wait_asynccnt)
  __builtin_amdgcn_s_wait_asynccnt(0);
#else
  asm volatile("s_wait_asynccnt 0x0" ::: "memory");
#endif
}

// ---------------- wave32 / block reductions --------------------------------
// Reduce TWO values with a single barrier sequence (3 barriers per pair).
template <bool MAXOP>
__device__ __forceinline__ float2 block_reduce2(float a, float b,
                                                float* s_ra, float* s_rb,
                                                float2* s_bc2) {
  const int lane = threadIdx.x & 31;
  const int wid  = threadIdx.x >> 5;   // 0..31 (blockDim.x == 1024)
#pragma unroll
  for (int o = 16; o > 0; o >>= 1) {
    float ta = __shfl_xor(a, o, 32);
    float tb = __shfl_xor(b, o, 32);
    a = MAXOP ? fmaxf(a, ta) : (a + ta);
    b = MAXOP ? fmaxf(b, tb) : (b + tb);
  }
  if (lane == 0) { s_ra[wid] = a; s_rb[wid] = b; }
  __syncthreads();
  if (wid == 0) {
    float xa = s_ra[lane];             // exactly 32 waves
    float xb = s_rb[lane];
#pragma unroll
    for (int o = 16; o > 0; o >>= 1) {
      float ta = __shfl_xor(xa, o, 32);
      float tb = __shfl_xor(xb, o, 32);
      xa = MAXOP ? fmaxf(xa, ta) : (xa + ta);
      xb = MAXOP ? fmaxf(xb, tb) : (xb + tb);
    }
    if (lane == 0) { s_bc2->x = xa; s_bc2->y = xb; }
  }
  __syncthreads();
  float2 r = *s_bc2;
  __syncthreads();                      // scratch reusable afterwards
  return r;
}

__device__ __forceinline__ float jsd_term(float pv, float qv,
                                          float lse_p, float lse_q) {
  float lp = pv - lse_p;
  float lq = qv - lse_q;
  float d  = lp - lq;
  return (__expf(lp) - __expf(lq)) * d;
}

// ---------------- per-row kernel: one HBM pass via 320KB LDS ---------------
__global__ void __launch_bounds__(NTHR, 1)
jsd_rows(const float* __restrict__ p, const float* __restrict__ q,
         const unsigned char* __restrict__ mask, float* __restrict__ row_kl) {
  __shared__ float4 s_p[V4];           // 128000 B
  __shared__ float4 s_q[V4];           // 128000 B   (total 250KB of 320KB LDS)
  __shared__ float  s_ra[NTHR / 32];
  __shared__ float  s_rb[NTHR / 32];
  __shared__ float2 s_bc2;

  const int tid = threadIdx.x;

  for (int row = blockIdx.x; row < NROWS; row += NBLK) {
    if (mask[row] == 0) {              // uniform branch: skip masked rows
      if (tid == 0) row_kl[row] = 0.0f;
      continue;
    }

    // Fire-and-forget L2 prefetch of the NEXT row this block will touch,
    // issued before the async burst so HBM streams it during our compute.
    {
      int nr = row + NBLK;
      if (nr < NROWS && mask[nr] != 0) {
        size_t off = (size_t)tid * 128;         // 1024 lanes x 128B >= 128000B
        if (off < (size_t)VDIM * 4) {
          __builtin_prefetch((const char*)(p + (size_t)nr * VDIM) + off, 0, 1);
          __builtin_prefetch((const char*)(q + (size_t)nr * VDIM) + off, 0, 1);
        }
      }
    }

    const float4* gp = (const float4*)p + (size_t)row * V4;
    const float4* gq = (const float4*)q + (size_t)row * V4;

    // Stage the whole row-pair into LDS with async loads (ASYNCcnt-tracked).
    for (int i = tid; i < V4; i += NTHR) {
      async_copy_b128(gp + i, &s_p[i]);
      async_copy_b128(gq + i, &s_q[i]);
    }

    wait_async_zero();
    __syncthreads();

    // Pass 1 (LDS): row maxima for stable logsumexp.
    float mp = -3.402823466e38f, mq = -3.402823466e38f;
    for (int i = tid; i < V4; i += NTHR) {
      float4 a = s_p[i];
      mp = fmaxf(mp, fmaxf(fmaxf(a.x, a.y), fmaxf(a.z, a.w)));
      float4 b = s_q[i];
      mq = fmaxf(mq, fmaxf(fmaxf(b.x, b.y), fmaxf(b.z, b.w)));
    }
    float2 mx = block_reduce2<true>(mp, mq, s_ra, s_rb, &s_bc2);
    mp = mx.x; mq = mx.y;

    // Pass 2 (LDS): sum of exponentials.
    float sp = 0.0f, sq = 0.0f;
    for (int i = tid; i < V4; i += NTHR) {
      float4 a = s_p[i];
      sp += __expf(a.x - mp) + __expf(a.y - mp) +
            __expf(a.z - mp) + __expf(a.w - mp);
      float4 b = s_q[i];
      sq += __expf(b.x - mq) + __expf(b.y - mq) +
            __expf(b.z - mq) + __expf(b.w - mq);
    }
    float2 sm = block_reduce2<false>(sp, sq, s_ra, s_rb, &s_bc2);
    const float lse_p = mp + __logf(sm.x);
    const float lse_q = mq + __logf(sm.y);

    // Pass 3 (LDS): row_kl = sum_v (P - Q) * (lp - lq).
    float acc = 0.0f;
    for (int i = tid; i < V4; i += NTHR) {
      float4 a = s_p[i];
      float4 b = s_q[i];
      acc += jsd_term(a.x, b.x, lse_p, lse_q);
      acc += jsd_term(a.y, b.y, lse_p, lse_q);
      acc += jsd_term(a.z, b.z, lse_p, lse_q);
      acc += jsd_term(a.w, b.w, lse_p, lse_q);
    }
    float2 ac = block_reduce2<false>(acc, 0.0f, s_ra, s_rb, &s_bc2);
    if (tid == 0) row_kl[row] = ac.x;
    // block_reduce2 ended with a barrier: safe to re-stage LDS next iteration.
  }
}

// ---------------- final deterministic scalar reduction ---------------------
__global__ void __launch_bounds__(NTHR, 1)
jsd_final(const float* __restrict__ row_kl, const unsigned char* __restrict__ mask,
          float* __restrict__ out) {
  __shared__ float  s_ra[NTHR / 32];
  __shared__ float  s_rb[NTHR / 32];
  __shared__ float2 s_bc2;
  float s = 0.0f, n = 0.0f;
  for (int i = threadIdx.x; i < NROWS; i += NTHR) {
    float w = (mask[i] != 0) ? 1.0f : 0.0f;
    s += row_kl[i] * w;
    n += w;
  }
  float2 r = block_reduce2<false>(s, n, s_ra, s_rb, &s_bc2);
  if (threadIdx.x == 0) out[0] = 0.25f * r.x / fmaxf(r.y, 1.0f);
}

extern "C" void kernel_launch(void* const* d_in, const int* in_sizes, int n_in,
                              void* d_out, int out_size, void* d_ws, size_t ws_size,
                              hipStream_t stream) {
  const float*         p    = (const float*)d_in[0];
  const float*         q    = (const float*)d_in[1];
  const unsigned char* mask = (const unsigned char*)d_in[2];
  float* row_kl = (float*)d_ws;                 // 4096 floats of scratch
  float* out    = (float*)d_out;

  jsd_rows<<<NBLK, NTHR, 0, stream>>>(p, q, mask, row_kl);
  jsd_final<<<1, NTHR, 0, stream>>>(row_kl, mask, out);

  (void)in_sizes; (void)n_in; (void)out_size; (void)ws_size;
}